// CA_34900904247668
// MI455X (gfx1250) — compile-verified
//
#include <hip/hip_runtime.h>

typedef __attribute__((ext_vector_type(2))) float v2f;
typedef __attribute__((ext_vector_type(4))) float v4f;
typedef __attribute__((ext_vector_type(8))) float v8f;

#define BN_EPS 1e-5f
#define B_  32
#define C_  256
#define CM_ 128
#define CO_ 256
#define H_  64
#define W_  64
#define L_  128   // H + W

// ---------------------------------------------------------------------------
// Kernel A: dual-axis mean pooling.
// x:(B,C,64,64) -> p:(B*C,128): p[.,0:64]=mean over w (xh), p[.,64:128]=mean over h (xw)
// One 256-thread block per (b,c) plane; LDS tile padded to stride 65 so both
// row-wise and column-wise reduction reads are bank-conflict-free.
// Regular (temporal) loads on purpose: x (134 MB) fits in the 192 MB L2 and
// is re-read by ca_scale.
// ---------------------------------------------------------------------------
__global__ __launch_bounds__(256) void ca_pool(const float* __restrict__ x,
                                               float* __restrict__ p) {
    __shared__ float tile[64 * 65];
    const int bc = blockIdx.x;
    const float* xp = x + (size_t)bc * (H_ * W_);
    const int t  = threadIdx.x;
    const int h  = t >> 2;
    const int w0 = (t & 3) * 16;

    const v4f* src = (const v4f*)(xp + h * W_ + w0);
#pragma unroll
    for (int j = 0; j < 4; ++j) {
        v4f v = src[j];
#pragma unroll
        for (int e = 0; e < 4; ++e)
            tile[h * 65 + w0 + j * 4 + e] = v[e];
    }
    __syncthreads();

    if (t < 64) {                       // xh[h] = mean over w
        float s = 0.0f;
#pragma unroll 8
        for (int w = 0; w < 64; ++w) s += tile[t * 65 + w];
        p[(size_t)bc * L_ + t] = s * (1.0f / 64.0f);
    } else if (t < 128) {               // xw[w] = mean over h
        const int w = t - 64;
        float s = 0.0f;
#pragma unroll 8
        for (int hh = 0; hh < 64; ++hh) s += tile[hh * 65 + w];
        p[(size_t)bc * L_ + 64 + w] = s * (1.0f / 64.0f);
    }
}

// ---------------------------------------------------------------------------
// Kernel B: per-batch MLP chain with fp32 WMMA (V_WMMA_F32_16X16X4_F32).
//   t  = hswish(bn(w1 @ p_b + b1))        (128 x 128)
//   g[:, :64]  = sigmoid(w2 @ t[:, :64]  + b2)
//   g[:, 64:]  = sigmoid(w3 @ t[:, 64:]  + b3)
// One block (8 waves) per batch. Wave-uniform control flow only (EXEC all-1s).
// ---------------------------------------------------------------------------
__global__ __launch_bounds__(256) void ca_mlp(
    const float* __restrict__ p,   // (B,256,128)
    const float* __restrict__ w1,  // (128,256)
    const float* __restrict__ b1,
    const float* __restrict__ gam, const float* __restrict__ bet,
    const float* __restrict__ mu,  const float* __restrict__ var,
    const float* __restrict__ w2,  const float* __restrict__ b2,  // (256,128)
    const float* __restrict__ w3,  const float* __restrict__ b3,  // (256,128)
    float* __restrict__ g)         // (B,256,128)
{
    __shared__ float lt[CM_ * L_];        // 128x128 f32 = 64 KB
    const int b     = blockIdx.x;
    const float* pb = p + (size_t)b * C_ * L_;
    const int wave  = threadIdx.x >> 5;
    const int lane  = threadIdx.x & 31;
    const int lrow  = lane & 15;          // N (or M/O) index within a 16-wide tile
    const int lhalf = lane >> 4;          // 0/1 half-lane group

    // ---------- phase 1: y = w1 @ pb, rows M0..M0+15 owned by this wave ----
    const int M0 = wave * 16;
    v8f acc[8];
#pragma unroll
    for (int n = 0; n < 8; ++n) acc[n] = (v8f)(0.0f);

    for (int k = 0; k < C_; k += 4) {
        // A frag (16x4 of w1): lane -> row M0+lrow, cols k+2*lhalf, +1
        const float* ar = w1 + (M0 + lrow) * C_ + k + 2 * lhalf;
        v2f a; a[0] = ar[0]; a[1] = ar[1];
#pragma unroll
        for (int n = 0; n < 8; ++n) {
            const int col = n * 16 + lrow;
            v2f bb;
            bb[0] = pb[(k     + 2 * lhalf) * L_ + col];
            bb[1] = pb[(k + 1 + 2 * lhalf) * L_ + col];
            acc[n] = __builtin_amdgcn_wmma_f32_16x16x4_f32(
                false, a, false, bb, (short)0, acc[n], false, false);
        }
    }

    // fused bias + BN + hswish epilogue; per-lane channel params (8 m values)
    float sc8[8], sh8[8];
#pragma unroll
    for (int i = 0; i < 8; ++i) {
        const int m = M0 + i + 8 * lhalf;
        const float s = gam[m] * rsqrtf(var[m] + BN_EPS);
        sc8[i] = s;
        sh8[i] = bet[m] + (b1[m] - mu[m]) * s;
    }
#pragma unroll
    for (int n = 0; n < 8; ++n) {
#pragma unroll
        for (int i = 0; i < 8; ++i) {
            const int m = M0 + i + 8 * lhalf;
            const float yv = acc[n][i] * sc8[i] + sh8[i];
            const float hs = yv * fminf(fmaxf(yv + 3.0f, 0.0f), 6.0f) * (1.0f / 6.0f);
            lt[m * L_ + n * 16 + lrow] = hs;
        }
    }
    __syncthreads();

    // ---------- phase 2: g = sigmoid(wX @ t + bX), 2 o-strips per wave -----
    float* gb = g + (size_t)b * CO_ * L_;
#pragma unroll
    for (int oo = 0; oo < 2; ++oo) {
        const int O0 = (wave * 2 + oo) * 16;
        for (int ltile = 0; ltile < 8; ++ltile) {          // wave-uniform
            const float* ws = (ltile < 4) ? w2 : w3;
            const float* bs = (ltile < 4) ? b2 : b3;
            v8f acc2 = (v8f)(0.0f);
            for (int k = 0; k < CM_; k += 4) {
                const float* ar = ws + (O0 + lrow) * CM_ + k + 2 * lhalf;
                v2f a; a[0] = ar[0]; a[1] = ar[1];
                const int col = ltile * 16 + lrow;
                v2f bb;
                bb[0] = lt[(k     + 2 * lhalf) * L_ + col];
                bb[1] = lt[(k + 1 + 2 * lhalf) * L_ + col];
                acc2 = __builtin_amdgcn_wmma_f32_16x16x4_f32(
                    false, a, false, bb, (short)0, acc2, false, false);
            }
#pragma unroll
            for (int i = 0; i < 8; ++i) {
                const int o  = O0 + i + 8 * lhalf;
                const float yv = acc2[i] + bs[o];
                const float gv = 1.0f / (1.0f + __expf(-yv));
                gb[o * L_ + ltile * 16 + lrow] = gv;
            }
        }
    }
}

// ---------------------------------------------------------------------------
// Kernel C: out = x * gh[b,c,h] * gw[b,c,w]. One block per (b,c) plane.
// x re-read should largely hit L2 (primed by ca_pool, 134 MB < 192 MB L2);
// output stores are non-temporal so the 134 MB write stream does not evict
// the L2-resident x while we are still consuming it.
// ---------------------------------------------------------------------------
__global__ __launch_bounds__(256) void ca_scale(const float* __restrict__ x,
                                                const float* __restrict__ g,
                                                float* __restrict__ out) {
    __shared__ float gr[L_];
    const int bc = blockIdx.x;
    const int t  = threadIdx.x;
    if (t < L_) gr[t] = g[(size_t)bc * L_ + t];
    __syncthreads();

    const float* xp = x   + (size_t)bc * (H_ * W_);
    float*       op = out + (size_t)bc * (H_ * W_);
    const int h  = t >> 2;
    const int w0 = (t & 3) * 16;
    const float gh = gr[h];

    const v4f* src = (const v4f*)(xp + h * W_ + w0);
    v4f*       dst = (v4f*)(op + h * W_ + w0);
#pragma unroll
    for (int j = 0; j < 4; ++j) {
        v4f v = src[j];
#pragma unroll
        for (int e = 0; e < 4; ++e)
            v[e] = v[e] * gh * gr[64 + w0 + j * 4 + e];
        __builtin_nontemporal_store(v, dst + j);   // TH=NT: keep x hot in L2
    }
}

// ---------------------------------------------------------------------------
extern "C" void kernel_launch(void* const* d_in, const int* in_sizes, int n_in,
                              void* d_out, int out_size, void* d_ws, size_t ws_size,
                              hipStream_t stream) {
    const float* x    = (const float*)d_in[0];
    const float* w1   = (const float*)d_in[1];
    const float* b1   = (const float*)d_in[2];
    const float* gam  = (const float*)d_in[3];
    const float* bet  = (const float*)d_in[4];
    const float* mu   = (const float*)d_in[5];
    const float* var  = (const float*)d_in[6];
    const float* w2   = (const float*)d_in[7];
    const float* b2   = (const float*)d_in[8];
    const float* w3   = (const float*)d_in[9];
    const float* b3   = (const float*)d_in[10];
    float* out = (float*)d_out;

    float* p = (float*)d_ws;                 // (B*C,128) pooled: 4 MB
    float* g = p + (size_t)B_ * C_ * L_;     // (B,256,128) gates: 4 MB

    ca_pool <<<B_ * C_, 256, 0, stream>>>(x, p);
    ca_mlp  <<<B_,       256, 0, stream>>>(p, w1, b1, gam, bet, mu, var,
                                           w2, b2, w3, b3, g);
    ca_scale<<<B_ * C_, 256, 0, stream>>>(x, g, out);
}